// SimpleGAT_51599737094672
// MI455X (gfx1250) — compile-verified
//
#include <hip/hip_runtime.h>
#include <hip/hip_bf16.h>

// ---------------------------------------------------------------------------
// SimpleGAT on MI455X (gfx1250).
// Structure exploited: equal-size graphs, dst = node repeated K times, so all
// segment ops are per-node loops over exactly K=7 neighbors (deterministic,
// no atomics). GEMMs use V_WMMA_F32_16X16X4_F32 (fp32 exact, CDNA5 WMMA path).
// ---------------------------------------------------------------------------

#define NN   131072   // total nodes
#define GB   256      // graphs
#define NPG  512      // nodes per graph
#define KNN  7        // neighbors
#define NH   4        // heads

typedef float v2f __attribute__((ext_vector_type(2)));
typedef float v8f __attribute__((ext_vector_type(8)));

// ---------------- workspace layout (bytes) ----------------
// srcIdx : NN*KNN int32                     =  3,670,016
// xl     : NN*128 f32 (reused both layers)  = 67,108,864
// xr     : NN*128 f32                       = 67,108,864
// h1     : NN*64  f32                       = 33,554,432
// h2     : NN*128 f32                       = 67,108,864
// total ~= 238.5 MB
static constexpr size_t OFF_SRC = 0;
static constexpr size_t SZ_SRC  = (size_t)NN * KNN * sizeof(int);
static constexpr size_t SZ_F128 = (size_t)NN * 128 * sizeof(float);
static constexpr size_t SZ_F64  = (size_t)NN * 64  * sizeof(float);
static constexpr size_t OFF_XL  = OFF_SRC + SZ_SRC;
static constexpr size_t OFF_XR  = OFF_XL + SZ_F128;
static constexpr size_t OFF_H1  = OFF_XR + SZ_F128;
static constexpr size_t OFF_H2  = OFF_H1 + SZ_F64;

// ---------------------------------------------------------------------------
// 1) Brute-force per-graph kNN. One block (256 threads) per graph; positions
//    staged in LDS (512*3*4 = 6 KB). Each thread keeps a sorted top-7.
// ---------------------------------------------------------------------------
__global__ void knn_kernel(const float* __restrict__ pos, int* __restrict__ srcIdx) {
  __shared__ float sp[NPG * 3];
  const int g = blockIdx.x;
  const int t = threadIdx.x;
  const float* gp = pos + (size_t)g * NPG * 3;
  for (int i = t; i < NPG * 3; i += blockDim.x) sp[i] = gp[i];
  __syncthreads();

  for (int n = t; n < NPG; n += blockDim.x) {
    const float px = sp[n * 3 + 0];
    const float py = sp[n * 3 + 1];
    const float pz = sp[n * 3 + 2];
    float bd[KNN];
    int   bi[KNN];
#pragma unroll
    for (int k = 0; k < KNN; ++k) { bd[k] = 3.0e38f; bi[k] = 0; }
    for (int j = 0; j < NPG; ++j) {
      const float dx = sp[j * 3 + 0] - px;
      const float dy = sp[j * 3 + 1] - py;
      const float dz = sp[j * 3 + 2] - pz;
      const float d2 = dx * dx + dy * dy + dz * dz;
      if (j == n) continue;                 // reference masks diagonal with 1e10
      if (d2 < bd[KNN - 1]) {               // insertion into ascending top-K
        int k = KNN - 1;
        while (k > 0 && bd[k - 1] > d2) {
          bd[k] = bd[k - 1]; bi[k] = bi[k - 1]; --k;
        }
        bd[k] = d2; bi[k] = j;
      }
    }
    const size_t base = ((size_t)g * NPG + n) * KNN;
#pragma unroll
    for (int k = 0; k < KNN; ++k) srcIdx[base + k] = g * NPG + bi[k];
  }
}

// ---------------------------------------------------------------------------
// 2) WMMA fp32 GEMM + bias:  C[M,ND] = A[M,KD] * W[KD,ND] + bias
//    One 16x16 output tile per wave via V_WMMA_F32_16X16X4_F32; 8 waves/block.
//    VGPR layouts per CDNA5 ISA 7.12.2:
//      A (16x4 f32): lane&15 = M row; half (lane>>4) selects K={0,1}/{2,3}
//      B (4x16 f32): lane&15 = N col; half selects K rows
//      C/D (16x16):  lane&15 = N col; VGPR v -> M = v + (lane>=16 ? 8 : 0)
// ---------------------------------------------------------------------------
template <int KD, int ND>
__global__ void wmma_gemm_bias(const float* __restrict__ A,
                               const float* __restrict__ W,
                               const float* __restrict__ bias,
                               float* __restrict__ C, int M) {
  const int lane  = threadIdx.x & 31;
  const int wave  = threadIdx.x >> 5;          // 0..7
  const int tile  = blockIdx.x * 8 + wave;
  const int ntn   = ND / 16;
  const int m0    = (tile / ntn) * 16;
  const int n0    = (tile % ntn) * 16;
  if (m0 >= M) return;

  const int row   = m0 + (lane & 15);          // A fragment row
  const int col   = n0 + (lane & 15);          // B/D fragment column
  const int khalf = (lane >> 4) * 2;           // 0 or 2

  v8f acc = {};
  constexpr int KP = (KD + 3) & ~3;            // pad K to multiple of 4
#pragma unroll
  for (int kk = 0; kk < KP; kk += 4) {
    const int ka = kk + khalf;
    v2f a, b;
    a.x = (ka     < KD) ? A[(size_t)row * KD + ka]     : 0.0f;
    a.y = (ka + 1 < KD) ? A[(size_t)row * KD + ka + 1] : 0.0f;
    b.x = (ka     < KD) ? W[(size_t)ka * ND + col]       : 0.0f;
    b.y = (ka + 1 < KD) ? W[(size_t)(ka + 1) * ND + col] : 0.0f;
    acc = __builtin_amdgcn_wmma_f32_16x16x4_f32(
        /*neg_a=*/false, a, /*neg_b=*/false, b,
        /*c_mod=*/(short)0, acc, /*reuse_a=*/false, /*reuse_b=*/false);
  }

  const float bv = bias[col];
  const int mr0 = m0 + ((lane >> 4) ? 8 : 0);
#pragma unroll
  for (int v = 0; v < 8; ++v) {
    C[(size_t)(mr0 + v) * ND + col] = acc[v] + bv;
  }
}

// ---------------------------------------------------------------------------
// 3) GATv2 edge + softmax + aggregate. One thread per (node, head).
//    alpha_j = sum_c leakyrelu(xl[s_j]+xr[i], 0.2) * att[h,c]
//    out[i]  = sum_j softmax_j(alpha) * xl[s_j]  (+ layer bias)
//    Feature arrays are L2-resident (<= 67 MB vs 192 MB L2).
// ---------------------------------------------------------------------------
template <int C>
__global__ void gat_kernel(const float* __restrict__ xl,
                           const float* __restrict__ xr,
                           const int* __restrict__ srcIdx,
                           const float* __restrict__ att,    // [NH, C]
                           const float* __restrict__ bias,   // [NH*C]
                           float* __restrict__ out) {
  const int tid  = blockIdx.x * blockDim.x + threadIdx.x;
  const int node = tid >> 2;           // NH == 4
  const int h    = tid & 3;
  if (node >= NN) return;
  constexpr int HC = NH * C;

  float xrv[C], attv[C];
#pragma unroll
  for (int c = 0; c < C; ++c) xrv[c]  = xr[(size_t)node * HC + h * C + c];
#pragma unroll
  for (int c = 0; c < C; ++c) attv[c] = att[h * C + c];

  int nbr[KNN];
#pragma unroll
  for (int k = 0; k < KNN; ++k) nbr[k] = srcIdx[(size_t)node * KNN + k];

  float alpha[KNN];
  float amax = -3.0e38f;
#pragma unroll
  for (int k = 0; k < KNN; ++k) {
    const float* xls = xl + (size_t)nbr[k] * HC + h * C;
    float s = 0.0f;
#pragma unroll
    for (int c = 0; c < C; ++c) {
      float v = xls[c] + xrv[c];
      v = (v > 0.0f) ? v : 0.2f * v;   // leaky_relu(., 0.2)
      s += v * attv[c];
    }
    alpha[k] = s;
    amax = fmaxf(amax, s);
  }

  float denom = 0.0f;
#pragma unroll
  for (int k = 0; k < KNN; ++k) { alpha[k] = expf(alpha[k] - amax); denom += alpha[k]; }
  const float inv = 1.0f / denom;

  float accv[C];
#pragma unroll
  for (int c = 0; c < C; ++c) accv[c] = 0.0f;
#pragma unroll
  for (int k = 0; k < KNN; ++k) {
    const float* xls = xl + (size_t)nbr[k] * HC + h * C;
    const float a = alpha[k] * inv;
#pragma unroll
    for (int c = 0; c < C; ++c) accv[c] += a * xls[c];
  }
#pragma unroll
  for (int c = 0; c < C; ++c)
    out[(size_t)node * HC + h * C + c] = accv[c] + bias[h * C + c];
}

// ---------------------------------------------------------------------------
// 4) Global mean pool: one block per graph, one thread per feature (128).
// ---------------------------------------------------------------------------
__global__ void pool_kernel(const float* __restrict__ h2, float* __restrict__ out) {
  const int b = blockIdx.x;
  const int f = threadIdx.x;           // 0..127
  const float* p = h2 + (size_t)b * NPG * 128 + f;
  float s = 0.0f;
  for (int n = 0; n < NPG; ++n) s += p[(size_t)n * 128];
  out[(size_t)b * 128 + f] = s * (1.0f / NPG);
}

// ---------------------------------------------------------------------------
extern "C" void kernel_launch(void* const* d_in, const int* in_sizes, int n_in,
                              void* d_out, int out_size, void* d_ws, size_t ws_size,
                              hipStream_t stream) {
  (void)in_sizes; (void)n_in; (void)out_size; (void)ws_size;

  const float* pos   = (const float*)d_in[0];
  const float* x     = (const float*)d_in[1];
  // d_in[2] = batch ids (int32) -- structure is implicit, unused
  const float* Wl1   = (const float*)d_in[3];
  const float* bl1   = (const float*)d_in[4];
  const float* Wr1   = (const float*)d_in[5];
  const float* br1   = (const float*)d_in[6];
  const float* att1  = (const float*)d_in[7];
  const float* bias1 = (const float*)d_in[8];
  const float* Wl2   = (const float*)d_in[9];
  const float* bl2   = (const float*)d_in[10];
  const float* Wr2   = (const float*)d_in[11];
  const float* br2   = (const float*)d_in[12];
  const float* att2  = (const float*)d_in[13];
  const float* bias2 = (const float*)d_in[14];
  float* out = (float*)d_out;

  char* ws = (char*)d_ws;
  int*   srcIdx = (int*)  (ws + OFF_SRC);
  float* xl     = (float*)(ws + OFF_XL);
  float* xr     = (float*)(ws + OFF_XR);
  float* h1     = (float*)(ws + OFF_H1);
  float* h2     = (float*)(ws + OFF_H2);

  // 1) kNN edges
  knn_kernel<<<GB, 256, 0, stream>>>(pos, srcIdx);

  // 2) layer-1 projections: [N,10] x [10,64]  (K padded 10->12)
  {
    const int tiles = (NN / 16) * (64 / 16);      // 32768
    wmma_gemm_bias<10, 64><<<tiles / 8, 256, 0, stream>>>(x, Wl1, bl1, xl, NN);
    wmma_gemm_bias<10, 64><<<tiles / 8, 256, 0, stream>>>(x, Wr1, br1, xr, NN);
  }

  // 3) layer-1 attention/aggregation -> h1 [N,64]
  gat_kernel<16><<<(NN * NH) / 256, 256, 0, stream>>>(xl, xr, srcIdx, att1, bias1, h1);

  // 4) layer-2 projections: [N,64] x [64,128]
  {
    const int tiles = (NN / 16) * (128 / 16);     // 65536
    wmma_gemm_bias<64, 128><<<tiles / 8, 256, 0, stream>>>(h1, Wl2, bl2, xl, NN);
    wmma_gemm_bias<64, 128><<<tiles / 8, 256, 0, stream>>>(h1, Wr2, br2, xr, NN);
  }

  // 5) layer-2 attention/aggregation -> h2 [N,128]
  gat_kernel<32><<<(NN * NH) / 256, 256, 0, stream>>>(xl, xr, srcIdx, att2, bias2, h2);

  // 6) per-graph mean pool -> [B,128]
  pool_kernel<<<GB, 128, 0, stream>>>(h2, out);
}